// IntraMetaPathAggregation_79594333929734
// MI455X (gfx1250) — compile-verified
//
#include <hip/hip_runtime.h>
#include <hip/hip_bf16.h>
#include <math.h>

#define D_DIM   128
#define H_HEADS 2
#define L_PATH  4

typedef __attribute__((ext_vector_type(2))) float v2f;
typedef __attribute__((ext_vector_type(8))) float v8f;

// ---------------------------------------------------------------------------
// Init: zero output accumulator + denom, set segment-max to -inf.
// ---------------------------------------------------------------------------
__global__ __launch_bounds__(256) void init_kernel(float* __restrict__ out,
                                                   float* __restrict__ segmax,
                                                   float* __restrict__ denom,
                                                   int n_nodes) {
    size_t i = (size_t)blockIdx.x * 256 + threadIdx.x;
    if (i < (size_t)n_nodes * (H_HEADS * D_DIM)) out[i] = 0.0f;
    if (i < (size_t)n_nodes * H_HEADS) {
        segmax[i] = -INFINITY;
        denom[i]  = 0.0f;
    }
}

// ---------------------------------------------------------------------------
// QKV projection: Y[w][n, 256] = X[n,128] @ W_w[128,256] for w in {q,k,v}.
// f32 WMMA 16x16x4, exact f32 accumulation. Block = 256 threads = 8 waves.
// Each block owns one 16-row X tile (staged in LDS, padded stride 132);
// wave w computes output column tiles {2w, 2w+1} (16 cols each) = all 256 cols.
// ---------------------------------------------------------------------------
__global__ __launch_bounds__(256) void qkv_gemm_wmma(const float* __restrict__ X,
                                                     const float* __restrict__ Wq,
                                                     const float* __restrict__ Wk,
                                                     const float* __restrict__ Wv,
                                                     float* __restrict__ Y,   // [3][N][256]
                                                     int n_nodes) {
    __shared__ float xs[16 * 132];
    const int tid = threadIdx.x;
    const int m0  = blockIdx.x * 16;
    const float* W = (blockIdx.y == 0) ? Wq : (blockIdx.y == 1) ? Wk : Wv;
    float* Yb = Y + (size_t)blockIdx.y * (size_t)n_nodes * 256;

    // Cooperative load of 16x128 f32 tile: 512 float4 by 256 threads.
    for (int i = tid; i < 16 * 32; i += 256) {
        int r = i >> 5, c4 = i & 31;
        int gr = m0 + r;
        float4 v = make_float4(0.f, 0.f, 0.f, 0.f);
        if (gr < n_nodes) v = ((const float4*)(X + (size_t)gr * D_DIM))[c4];
        ((float4*)(xs + r * 132))[c4] = v;
    }
    __syncthreads();

    const int lane = tid & 31;
    const int wave = tid >> 5;
    const int mrow = lane & 15;
    const int koff = (lane >> 4) << 1;   // 0 (lanes 0-15) or 2 (lanes 16-31)
    const int ncol = lane & 15;

    for (int nt = 0; nt < 2; ++nt) {
        const int n0 = (wave * 2 + nt) * 16;
        v8f acc = {0.f, 0.f, 0.f, 0.f, 0.f, 0.f, 0.f, 0.f};
#pragma unroll
        for (int k0 = 0; k0 < D_DIM; k0 += 4) {
            v2f a, b;
            // A 16x4 f32: lane = M, VGPR0 holds K=koff, VGPR1 holds K=koff+1
            a.x = xs[mrow * 132 + k0 + koff];
            a.y = xs[mrow * 132 + k0 + koff + 1];
            // B 4x16 f32 (K x N): lane = N, split K across lane halves
            const float* wp = W + (size_t)(k0 + koff) * 256 + n0 + ncol;
            b.x = wp[0];
            b.y = wp[256];
            acc = __builtin_amdgcn_wmma_f32_16x16x4_f32(false, a, false, b,
                                                        (short)0, acc, false, false);
        }
        // C/D layout: VGPR r -> M = r + (lane<16 ? 0 : 8), N = lane&15
        const int col   = n0 + ncol;
        const int rbase = m0 + ((lane >> 4) << 3);
#pragma unroll
        for (int r = 0; r < 8; ++r) {
            int rr = rbase + r;
            if (rr < n_nodes) Yb[(size_t)rr * 256 + col] = acc[r];
        }
    }
}

// ---------------------------------------------------------------------------
// CAS-based float atomic max (init value is -inf).
// ---------------------------------------------------------------------------
__device__ inline void atomic_max_f32(float* addr, float val) {
    int* ai  = (int*)addr;
    int  old = __float_as_int(*addr);
    while (__int_as_float(old) < val) {
        int prev = atomicCAS(ai, old, __float_as_int(val));
        if (prev == old) break;
        old = prev;
    }
}

// ---------------------------------------------------------------------------
// Pass 1: wave per (edge, head). Gather metapath rows (coalesced float4 per
// lane), mean -> agg; logit = q[dst] . (k[src]+agg) / sqrt(D); seg-max.
// ---------------------------------------------------------------------------
__global__ __launch_bounds__(256) void edge_logits_kernel(
        const float* __restrict__ nf, const int* __restrict__ ei,
        const int* __restrict__ mp, const float* __restrict__ qkv,
        float* __restrict__ logits, float* __restrict__ segmax,
        int n_nodes, int n_edges) {
    int gw   = (blockIdx.x * 256 + threadIdx.x) >> 5;
    int lane = threadIdx.x & 31;
    if (gw >= n_edges * H_HEADS) return;
    int e = gw >> 1, h = gw & 1;
    int src = ei[e], dst = ei[n_edges + e];

    float ax = 0.f, ay = 0.f, az = 0.f, aw = 0.f;
#pragma unroll
    for (int l = 0; l < L_PATH; ++l) {
        int nid = mp[e * L_PATH + l];
        const float4 f = ((const float4*)(nf + (size_t)nid * D_DIM))[lane];
        ax += f.x; ay += f.y; az += f.z; aw += f.w;
    }
    ax *= 0.25f; ay *= 0.25f; az *= 0.25f; aw *= 0.25f;

    const float* qp = qkv + (size_t)dst * 256 + h * D_DIM;                          // Q
    const float* kp = qkv + (size_t)n_nodes * 256 + (size_t)src * 256 + h * D_DIM;  // K
    float4 q4 = ((const float4*)qp)[lane];
    float4 k4 = ((const float4*)kp)[lane];
    float part = q4.x * (k4.x + ax) + q4.y * (k4.y + ay) +
                 q4.z * (k4.z + az) + q4.w * (k4.w + aw);
#pragma unroll
    for (int off = 16; off > 0; off >>= 1) part += __shfl_down(part, off, 32);
    if (lane == 0) {
        float logit = part * 0.08838834764831845f;   // 1/sqrt(128)
        logits[gw] = logit;
        atomic_max_f32(&segmax[dst * H_HEADS + h], logit);
    }
}

// ---------------------------------------------------------------------------
// Pass 2: ex = exp(logit - m[dst]); denom[dst] += ex.
// ---------------------------------------------------------------------------
__global__ __launch_bounds__(256) void edge_exp_kernel(
        const int* __restrict__ ei, float* __restrict__ logits,
        const float* __restrict__ segmax, float* __restrict__ denom,
        int n_edges) {
    int i = blockIdx.x * 256 + threadIdx.x;
    if (i >= n_edges * H_HEADS) return;
    int e = i >> 1, h = i & 1;
    int dst = ei[n_edges + e];
    float m = segmax[dst * H_HEADS + h];
    if (!(m > -3.0e38f && m < 3.0e38f)) m = 0.0f;   // match reference isfinite guard
    float ex = expf(logits[i] - m);
    logits[i] = ex;
    atomicAdd(&denom[dst * H_HEADS + h], ex);
}

// ---------------------------------------------------------------------------
// Pass 3: wave per (edge, head): out[dst] += ex * (v[src] + agg).
// ---------------------------------------------------------------------------
__global__ __launch_bounds__(256) void edge_scatter_kernel(
        const float* __restrict__ nf, const int* __restrict__ ei,
        const int* __restrict__ mp, const float* __restrict__ qkv,
        const float* __restrict__ ex_buf, float* __restrict__ out,
        int n_nodes, int n_edges) {
    int gw   = (blockIdx.x * 256 + threadIdx.x) >> 5;
    int lane = threadIdx.x & 31;
    if (gw >= n_edges * H_HEADS) return;
    int e = gw >> 1, h = gw & 1;
    int src = ei[e], dst = ei[n_edges + e];

    float ax = 0.f, ay = 0.f, az = 0.f, aw = 0.f;
#pragma unroll
    for (int l = 0; l < L_PATH; ++l) {
        int nid = mp[e * L_PATH + l];
        const float4 f = ((const float4*)(nf + (size_t)nid * D_DIM))[lane];
        ax += f.x; ay += f.y; az += f.z; aw += f.w;
    }
    ax *= 0.25f; ay *= 0.25f; az *= 0.25f; aw *= 0.25f;

    const float* vp = qkv + 2 * (size_t)n_nodes * 256 + (size_t)src * 256 + h * D_DIM;
    float4 v4 = ((const float4*)vp)[lane];
    float w  = ex_buf[gw];
    float* op = out + (size_t)dst * 256 + h * D_DIM + lane * 4;
    atomicAdd(op + 0, w * (v4.x + ax));
    atomicAdd(op + 1, w * (v4.y + ay));
    atomicAdd(op + 2, w * (v4.z + az));
    atomicAdd(op + 3, w * (v4.w + aw));
}

// ---------------------------------------------------------------------------
// Finalize: out /= denom + 1e-16.
// ---------------------------------------------------------------------------
__global__ __launch_bounds__(256) void finalize_kernel(float* __restrict__ out,
                                                       const float* __restrict__ denom,
                                                       int n_nodes) {
    size_t i = (size_t)blockIdx.x * 256 + threadIdx.x;
    if (i >= (size_t)n_nodes * 256) return;
    int n = (int)(i >> 8);
    int h = (int)((i >> 7) & 1);
    out[i] = out[i] / (denom[n * H_HEADS + h] + 1e-16f);
}

// ---------------------------------------------------------------------------
extern "C" void kernel_launch(void* const* d_in, const int* in_sizes, int n_in,
                              void* d_out, int out_size, void* d_ws, size_t ws_size,
                              hipStream_t stream) {
    const float* nf = (const float*)d_in[0];
    const int*   ei = (const int*)d_in[1];
    const int*   mp = (const int*)d_in[2];
    const float* Wq = (const float*)d_in[3];
    const float* Wk = (const float*)d_in[4];
    const float* Wv = (const float*)d_in[5];
    float* out = (float*)d_out;

    const int n_nodes = in_sizes[0] / D_DIM;
    const int n_edges = in_sizes[1] / 2;

    // Workspace layout (floats): QKV [3][N][256] | ex/logits [E*2] | segmax [N*2] | denom [N*2]
    float* ws     = (float*)d_ws;
    float* qkv    = ws;
    float* logits = qkv + (size_t)3 * n_nodes * 256;
    float* segmax = logits + (size_t)n_edges * H_HEADS;
    float* denom  = segmax + (size_t)n_nodes * H_HEADS;

    {   // init
        size_t tot = (size_t)n_nodes * 256;
        init_kernel<<<dim3((unsigned)((tot + 255) / 256)), 256, 0, stream>>>(out, segmax, denom, n_nodes);
    }
    {   // WMMA QKV projection
        dim3 grid((n_nodes + 15) / 16, 3);
        qkv_gemm_wmma<<<grid, 256, 0, stream>>>(nf, Wq, Wk, Wv, qkv, n_nodes);
    }
    {   // logits + segment max
        long waves = (long)n_edges * H_HEADS;
        edge_logits_kernel<<<dim3((unsigned)((waves * 32 + 255) / 256)), 256, 0, stream>>>(
            nf, ei, mp, qkv, logits, segmax, n_nodes, n_edges);
    }
    {   // exp + denom
        long items = (long)n_edges * H_HEADS;
        edge_exp_kernel<<<dim3((unsigned)((items + 255) / 256)), 256, 0, stream>>>(
            ei, logits, segmax, denom, n_edges);
    }
    {   // weighted scatter
        long waves = (long)n_edges * H_HEADS;
        edge_scatter_kernel<<<dim3((unsigned)((waves * 32 + 255) / 256)), 256, 0, stream>>>(
            nf, ei, mp, qkv, logits, out, n_nodes, n_edges);
    }
    {   // normalize
        size_t tot = (size_t)n_nodes * 256;
        finalize_kernel<<<dim3((unsigned)((tot + 255) / 256)), 256, 0, stream>>>(out, denom, n_nodes);
    }
}